// GNN_5368709120103
// MI455X (gfx1250) — compile-verified
//
#include <hip/hip_runtime.h>
#include <hip/hip_bf16.h>

typedef __attribute__((ext_vector_type(2))) float v2f;
typedef __attribute__((ext_vector_type(8))) float v8f;

#define DIM 128
#define BN_EPS 1e-5f

// XOR-swizzled transposed LDS index for W: element (k, n) of W[k*128+n].
// Pairs (k even, k odd) stay contiguous -> single ds_load_b64 per B fragment.
// Swizzle spreads the two half-waves over disjoint bank sets (conflict-free).
__device__ __forceinline__ int wswz(int k, int n)
{
    int p = (k >> 1) ^ ((n & 15) << 2) ^ ((n >> 4) & 3);
    return n * DIM + (p << 1) + (k & 1);
}

// ---------------------------------------------------------------------------
// h[n,:] = emb_atom[x[n,0],:] + emb_chir[x[n,1],:]
// ---------------------------------------------------------------------------
__global__ __launch_bounds__(256) void gin_embed(
    const int* __restrict__ x, const float* __restrict__ emb_atom,
    const float* __restrict__ emb_chir, float* __restrict__ h, int n_nodes)
{
    int idx = blockIdx.x * 256 + threadIdx.x;
    int node = idx >> 5;
    if (node >= n_nodes) return;
    int c = (idx & 31) * 4;
    int a  = x[node * 2 + 0];
    int ch = x[node * 2 + 1];
    const float4 va = *(const float4*)(emb_atom + (size_t)a  * DIM + c);
    const float4 vc = *(const float4*)(emb_chir + (size_t)ch * DIM + c);
    float4 o;
    o.x = va.x + vc.x; o.y = va.y + vc.y; o.z = va.z + vc.z; o.w = va.w + vc.w;
    *(float4*)(h + (size_t)node * DIM + c) = o;
}

// ---------------------------------------------------------------------------
// z[dst[e],:] += h[src[e],:]   (z pre-initialized with h  => h + agg)
// ---------------------------------------------------------------------------
__global__ __launch_bounds__(256) void gin_scatter_add(
    const float* __restrict__ h, const int* __restrict__ src,
    const int* __restrict__ dst, float* __restrict__ z, int n_edges)
{
    int e = blockIdx.x * 8 + (threadIdx.x >> 5);
    if (e >= n_edges) return;
    int lane = threadIdx.x & 31;
    const float* hs = h + (size_t)src[e] * DIM;
    float*       zd = z + (size_t)dst[e] * DIM;
#pragma unroll
    for (int i = 0; i < 4; ++i)
        atomicAdd(&zd[lane + 32 * i], hs[lane + 32 * i]);
}

// ---------------------------------------------------------------------------
// Out = act(A @ W + bias)    A:[N,128] W:[128,128]
// EPI==0: ReLU epilogue. EPI==1: identity + fused per-column sum/sumsq.
//
// 256 threads = 8 wave32s; wave computes 16 rows x 128 cols with
// v_wmma_f32_16x16x4_f32 (32 K-steps x 8 N-tiles).
//  - A panel (16x128) preloaded to registers: 32 back-to-back global b64 loads
//  - W transposed+swizzled in 64KB LDS: one conflict-free ds_load_b64 per frag
//
// ISA VGPR layouts (CDNA5 ISA):
//   A 16x4 f32 : lane L: M=L%16, VGPR r holds K = (L/16)*2 + r
//   B 4x16 f32 : lane L: N=L%16, VGPR r holds K = (L/16)*2 + r
//   C/D 16x16  : lane L: N=L%16, VGPR r holds M = r + 8*(L/16)
// ---------------------------------------------------------------------------
template <int EPI>
__global__ __launch_bounds__(256) void gin_gemm128(
    const float* __restrict__ A, const float* __restrict__ W,
    const float* __restrict__ bias, float* __restrict__ Out,
    float* __restrict__ col_sum, float* __restrict__ col_sumsq)
{
    __shared__ float wlds[DIM * DIM];   // 64 KB, transposed + swizzled
    int tid = threadIdx.x;
#pragma unroll 8
    for (int i = tid; i < DIM * DIM; i += 256) {
        int k = i >> 7, n = i & 127;     // consecutive tid -> consecutive n
        wlds[wswz(k, n)] = W[i];
    }
    __syncthreads();

    const int lane = tid & 31;
    const int wave = tid >> 5;
    const int m16  = lane & 15;          // N (B/C/D) or M (A) sub-index
    const int hi   = lane >> 4;          // half-wave: K-pair selector
    const size_t rowBase = (size_t)blockIdx.x * 128 + (size_t)wave * 16;

    // ---- preload full A panel: lane's K pairs for all 32 k-steps ----------
    const float* arow = A + (rowBase + m16) * DIM + hi * 2;
    v2f afrag[32];
#pragma unroll
    for (int k = 0; k < 32; ++k)
        afrag[k] = *(const v2f*)(arow + k * 4);   // K = 4k + 2*hi + {0,1}

    // ---- accumulators initialized with bias (depends only on N -> lane) --
    v8f acc[8];
#pragma unroll
    for (int nt = 0; nt < 8; ++nt) {
        float bv = bias[nt * 16 + m16];
        acc[nt] = (v8f){bv, bv, bv, bv, bv, bv, bv, bv};
    }

    // ---- main loop: 8 ds_load_b64 + 8 WMMA per k-step ---------------------
#pragma unroll 4
    for (int k = 0; k < 32; ++k) {
        const int kk = k * 4 + hi * 2;   // even
#pragma unroll
        for (int nt = 0; nt < 8; ++nt) {
            v2f b = *(const v2f*)&wlds[wswz(kk, nt * 16 + m16)];
            acc[nt] = __builtin_amdgcn_wmma_f32_16x16x4_f32(
                false, afrag[k], false, b, (short)0, acc[nt], false, false);
        }
    }

    // ---- epilogue ---------------------------------------------------------
#pragma unroll
    for (int nt = 0; nt < 8; ++nt) {
        float s = 0.f, ss = 0.f;
#pragma unroll
        for (int r = 0; r < 8; ++r) {
            float v = acc[nt][r];
            if (EPI == 0) v = fmaxf(v, 0.f);
            Out[(rowBase + r + 8 * hi) * DIM + nt * 16 + m16] = v;
            if (EPI == 1) { s += v; ss += v * v; }
        }
        if (EPI == 1) {
            s  += __shfl_down(s, 16, 32);     // lanes L and L+16: same column
            ss += __shfl_down(ss, 16, 32);
            if (hi == 0) {
                atomicAdd(&col_sum[nt * 16 + m16], s);
                atomicAdd(&col_sumsq[nt * 16 + m16], ss);
            }
        }
    }
}

// ---------------------------------------------------------------------------
__global__ void gin_zero_stats(float* __restrict__ stats)
{
    stats[threadIdx.x] = 0.f;   // 256 threads: sum[128] + sumsq[128]
}

__global__ void gin_bn_finalize(
    const float* __restrict__ col_sum, const float* __restrict__ col_sumsq,
    const float* __restrict__ gamma, const float* __restrict__ beta,
    float* __restrict__ scale, float* __restrict__ shift, float invN)
{
    int c = threadIdx.x;
    float mean = col_sum[c] * invN;
    float var  = col_sumsq[c] * invN - mean * mean;
    float sc   = gamma[c] * rsqrtf(var + BN_EPS);
    scale[c] = sc;
    shift[c] = beta[c] - mean * sc;
}

__global__ __launch_bounds__(256) void gin_bn_apply(
    const float4* __restrict__ u, const float* __restrict__ scale,
    const float* __restrict__ shift, float4* __restrict__ out,
    int n4, int do_relu)
{
    int idx = blockIdx.x * 256 + threadIdx.x;
    if (idx >= n4) return;
    int c = (idx & 31) * 4;
    float4 v = u[idx];
    float4 o;
    o.x = v.x * scale[c + 0] + shift[c + 0];
    o.y = v.y * scale[c + 1] + shift[c + 1];
    o.z = v.z * scale[c + 2] + shift[c + 2];
    o.w = v.w * scale[c + 3] + shift[c + 3];
    if (do_relu) {
        o.x = fmaxf(o.x, 0.f); o.y = fmaxf(o.y, 0.f);
        o.z = fmaxf(o.z, 0.f); o.w = fmaxf(o.w, 0.f);
    }
    out[idx] = o;
}

// ---------------------------------------------------------------------------
extern "C" void kernel_launch(void* const* d_in, const int* in_sizes, int n_in,
                              void* d_out, int out_size, void* d_ws, size_t ws_size,
                              hipStream_t stream)
{
    const int*   x        = (const int*)d_in[0];
    const int*   ei       = (const int*)d_in[1];   // [2, E]
    // d_in[2] = edge_attr (unused by reference math)
    const float* emb_atom = (const float*)d_in[3];
    const float* emb_chir = (const float*)d_in[4];
    const float* W1 = (const float*)d_in[5];
    const float* b1 = (const float*)d_in[6];
    const float* W2 = (const float*)d_in[7];
    const float* b2 = (const float*)d_in[8];
    const float* gamma = (const float*)d_in[9];
    const float* beta  = (const float*)d_in[10];

    const int N = in_sizes[0] / 2;      // 400000
    const int E = in_sizes[1] / 2;      // 800000
    const size_t HSZ = (size_t)N * DIM * sizeof(float);

    // Buffer A = workspace (h / t), Buffer B = d_out (z / u / final h)
    float* bufA  = (float*)d_ws;
    float* bufB  = (float*)d_out;
    float* stats = (float*)((char*)d_ws + HSZ);
    float* sum   = stats;
    float* sumsq = stats + 128;
    float* scale = stats + 256;
    float* shift = stats + 384;

    const int* src = ei;
    const int* dst = ei + E;

    const int n4       = N * (DIM / 4);
    const int gridElem = (n4 + 255) / 256;
    const int gridEdge = (E + 7) / 8;
    const int gridGemm = N / 128;                    // 400000/128 = 3125 exact

    gin_embed<<<gridElem, 256, 0, stream>>>(x, emb_atom, emb_chir, bufA, N);

    for (int l = 0; l < 5; ++l) {
        // z = h (bufB), then z[dst] += h[src]
        hipMemcpyAsync(bufB, bufA, HSZ, hipMemcpyDeviceToDevice, stream);
        gin_scatter_add<<<gridEdge, 256, 0, stream>>>(bufA, src, dst, bufB, E);

        // t = relu(z @ W1 + b1)   (bufB -> bufA)
        gin_gemm128<0><<<gridGemm, 256, 0, stream>>>(
            bufB, W1 + (size_t)l * DIM * DIM, b1 + l * DIM, bufA,
            nullptr, nullptr);

        // u = t @ W2 + b2  with fused column stats   (bufA -> bufB)
        gin_zero_stats<<<1, 256, 0, stream>>>(stats);
        gin_gemm128<1><<<gridGemm, 256, 0, stream>>>(
            bufA, W2 + (size_t)l * DIM * DIM, b2 + l * DIM, bufB,
            sum, sumsq);

        // batch-norm scale/shift, then apply (+ReLU if not last layer)
        gin_bn_finalize<<<1, 128, 0, stream>>>(
            sum, sumsq, gamma + l * DIM, beta + l * DIM, scale, shift,
            1.0f / (float)N);

        float* dstbuf = (l < 4) ? bufA : bufB;   // last layer: in-place d_out
        gin_bn_apply<<<gridElem, 256, 0, stream>>>(
            (const float4*)bufB, scale, shift, (float4*)dstbuf, n4,
            (l < 4) ? 1 : 0);
    }
}